// SpatialTransformer_range_flow_flow_list2_Multi_Flow_85555748536626
// MI455X (gfx1250) — compile-verified
//
#include <hip/hip_runtime.h>

// Volume geometry (fixed by the reference).
constexpr int Dz  = 96;
constexpr int Hy  = 160;
constexpr int Wx  = 160;
constexpr int HW  = Hy * Wx;        // 25600
constexpr int DHW = Dz * Hy * Wx;   // 2457600

// Combined normalize_grid + grid_sample unnormalize:
//   pos = v * S/(S-1) - 0.5   with v = voxel + dvf*range_flow
constexpr float SZ = (float)Dz / (float)(Dz - 1);
constexpr float SY = (float)Hy / (float)(Hy - 1);
constexpr float SX = (float)Wx / (float)(Wx - 1);

// 8-byte pair with only 4-byte alignment guarantee (x taps are at arbitrary
// float offsets). gfx1250 handles misaligned b64 VMEM in unaligned mode.
struct __attribute__((packed, aligned(4))) f2u {
  float a, b;
};

// Trilinear gather of NC planar channels (channel stride DHW) at (x,y,z),
// zeros padding. The two x-taps of each corner are adjacent in memory, so
// each (z,y) corner row is fetched as ONE b64 pair load at
// xc = clamp(x0, 0, W-2); boundary cases are handled branchlessly by swapping
// the two x-weights when the clamp shifted the base:
//   x0 == -1  : pair = (f[0]=f[x1], f[1]), u0 = 0  -> swapped = p.a*u1      OK
//   x0 == W-1 : pair = (f[W-2], f[W-1]=f[x0]), u1=0 -> swapped = p.b*u0    OK
//   fully OOB : both weights already 0                                      OK
template <int NC>
__device__ __forceinline__ void trilinear_gather(const float* __restrict__ f,
                                                 float x, float y, float z,
                                                 float* __restrict__ acc) {
  const float fx = floorf(x), fy = floorf(y), fz = floorf(z);
  const int x0 = (int)fx, y0 = (int)fy, z0 = (int)fz;
  const float tx = x - fx, ty = y - fy, tz = z - fz;

  // x: one clamped pair base + weight pair (validity folded in, swap on clamp)
  const int xc = min(max(x0, 0), Wx - 2);
  const float u0 = (x0 >= 0 && x0 < Wx)         ? 1.0f - tx : 0.0f;
  const float u1 = (x0 + 1 >= 0 && x0 + 1 < Wx) ? tx        : 0.0f;
  const bool swap_x = (x0 != xc);
  const float wa = swap_x ? u1 : u0;
  const float wb = swap_x ? u0 : u1;

  const int ys[2] = { min(max(y0, 0), Hy - 1), min(max(y0 + 1, 0), Hy - 1) };
  const int zs[2] = { min(max(z0, 0), Dz - 1), min(max(z0 + 1, 0), Dz - 1) };
  const float wyv[2] = { (y0 >= 0 && y0 < Hy)         ? 1.0f - ty : 0.0f,
                         (y0 + 1 >= 0 && y0 + 1 < Hy) ? ty        : 0.0f };
  const float wzv[2] = { (z0 >= 0 && z0 < Dz)         ? 1.0f - tz : 0.0f,
                         (z0 + 1 >= 0 && z0 + 1 < Dz) ? tz        : 0.0f };

#pragma unroll
  for (int c = 0; c < NC; ++c) acc[c] = 0.0f;

#pragma unroll
  for (int iz = 0; iz < 2; ++iz) {
#pragma unroll
    for (int iy = 0; iy < 2; ++iy) {
      const float wzy = wzv[iz] * wyv[iy];
      const float wA = wzy * wa;
      const float wB = wzy * wb;
      const int idx = zs[iz] * HW + ys[iy] * Wx + xc;
#pragma unroll
      for (int c = 0; c < NC; ++c) {
        const f2u p = *(const f2u*)(f + idx + c * DHW);
        acc[c] = fmaf(p.a, wA, fmaf(p.b, wB, acc[c]));
      }
    }
  }
}

// One flow-composition step:
//   comp_out[c] = trilinear_sample(comp_in[c], pos(dvf)) + dvf[c]
// comp_in, dvf, comp_out are [3, D, H, W] planar.
__global__ __launch_bounds__(256) void compose_flow_kernel(
    const float* __restrict__ comp_in,
    const float* __restrict__ dvf,
    const float* __restrict__ rf_ptr,
    float* __restrict__ comp_out) {
  const int i = blockIdx.x * blockDim.x + threadIdx.x;
  if (i >= DHW) return;

  // gfx1250 global_prefetch_b8: warm the gather neighborhood (taps land
  // within a few voxels of the identity position) while VALU computes taps.
  __builtin_prefetch(comp_in + i, 0, 3);

  const float rf = rf_ptr[0];  // uniform -> scalar load

  const int w = i % Wx;
  const int t = i / Wx;
  const int h = t % Hy;
  const int d = t / Hy;

  const float fd = dvf[i];
  const float fh = dvf[i + DHW];
  const float fw = dvf[i + 2 * DHW];

  const float z = fmaf((float)d + fd * rf, SZ, -0.5f);
  const float y = fmaf((float)h + fh * rf, SY, -0.5f);
  const float x = fmaf((float)w + fw * rf, SX, -0.5f);

  float acc[3];
  trilinear_gather<3>(comp_in, x, y, z, acc);

  comp_out[i]           = acc[0] + fd;
  comp_out[i + DHW]     = acc[1] + fh;
  comp_out[i + 2 * DHW] = acc[2] + fw;
}

// Final warp: deform[i] = trilinear_sample(src, pos(composed_flow)).
__global__ __launch_bounds__(256) void warp_image_kernel(
    const float* __restrict__ src,
    const float* __restrict__ flow,
    const float* __restrict__ rf_ptr,
    float* __restrict__ out) {
  const int i = blockIdx.x * blockDim.x + threadIdx.x;
  if (i >= DHW) return;

  __builtin_prefetch(src + i, 0, 3);

  const float rf = rf_ptr[0];

  const int w = i % Wx;
  const int t = i / Wx;
  const int h = t % Hy;
  const int d = t / Hy;

  const float fd = flow[i];
  const float fh = flow[i + DHW];
  const float fw = flow[i + 2 * DHW];

  const float z = fmaf((float)d + fd * rf, SZ, -0.5f);
  const float y = fmaf((float)h + fh * rf, SY, -0.5f);
  const float x = fmaf((float)w + fw * rf, SX, -0.5f);

  float acc[1];
  trilinear_gather<1>(src, x, y, z, acc);
  out[i] = acc[0];
}

extern "C" void kernel_launch(void* const* d_in, const int* in_sizes, int n_in,
                              void* d_out, int out_size, void* d_ws, size_t ws_size,
                              hipStream_t stream) {
  // setup_inputs order: src, flow_list, final_flow, range_flow (all fp32).
  const float* src        = (const float*)d_in[0];
  const float* flow_list  = (const float*)d_in[1];  // [3,1,3,D,H,W]
  const float* final_flow = (const float*)d_in[2];  // [1,3,D,H,W]
  const float* rf         = (const float*)d_in[3];  // scalar

  // d_out = concat(deform_2_img [DHW], out_flow [3*DHW]).
  float* deform   = (float*)d_out;
  float* out_flow = (float*)d_out + DHW;
  float* scratch  = (float*)d_ws;  // needs 3*DHW floats (~29.5 MB)

  const float* f0 = flow_list;            // flow_list[0]
  const float* f1 = flow_list + 3 * DHW;  // flow_list[1]
  const float* f2 = flow_list + 6 * DHW;  // flow_list[2]

  const int threads = 256;                            // 8 wave32 waves / block
  const int blocks  = (DHW + threads - 1) / threads;  // 9600 blocks

  // c1 = warp(f0 by f1) + f1        -> out_flow region (temp use)
  compose_flow_kernel<<<blocks, threads, 0, stream>>>(f0, f1, rf, out_flow);
  // c2 = warp(c1 by f2) + f2        -> scratch
  compose_flow_kernel<<<blocks, threads, 0, stream>>>(out_flow, f2, rf, scratch);
  // c3 = warp(c2 by final) + final  -> out_flow  (this IS output #2)
  compose_flow_kernel<<<blocks, threads, 0, stream>>>(scratch, final_flow, rf, out_flow);
  // deform_2_img = warp(src by c3)  -> output #1
  warp_image_kernel<<<blocks, threads, 0, stream>>>(src, out_flow, rf, deform);
}